// Gemma3nAudioCumulativeGroupNorm_12412455485700
// MI455X (gfx1250) — compile-verified
//
#include <hip/hip_runtime.h>

typedef float __attribute__((ext_vector_type(4))) vfloat4;

#define B_LEN   8
#define T_LEN   2048
#define N_RED   4096        // F*C elements reduced per (b,t)
#define EPS_GN  1e-3f

// ---------------------------------------------------------------------------
// Kernel 1: per-(b,t) sum & sum-of-squares.
// 16 KB row tile staged into LDS with CDNA5 async-tensor path
// (global_load_async_to_lds_b128 + s_wait_asynccnt), then wave32 shuffle
// reduction + tiny LDS cross-wave combine.
// ---------------------------------------------------------------------------
__global__ __launch_bounds__(256) void k_stats(const float* __restrict__ x,
                                               float2* __restrict__ stats) {
    __shared__ vfloat4 tile[1024];          // 16 KB: one (b,t) row
    __shared__ float ps1[8];
    __shared__ float ps2[8];

    const int bt  = blockIdx.x;             // 0 .. B*T-1
    const int tid = threadIdx.x;
    const float* rowbase = x + (size_t)bt * N_RED;   // uniform -> SGPR pair

    // Async copy 16 KB tile: each lane issues 4 x 16B, GVS addressing
    // (saddr = row base SGPRs, vaddr = 32-bit byte offset VGPR,
    //  vdst = LDS byte address VGPR).  Tracked on ASYNCcnt.
#pragma unroll
    for (int k = 0; k < 4; ++k) {
        const int idx     = tid + k * 256;                    // float4 index
        unsigned lds_addr = (unsigned)(size_t)(&tile[idx]);   // LDS byte addr (low 32 of generic)
        unsigned goff     = (unsigned)idx * 16u;              // global byte offset
        asm volatile("global_load_async_to_lds_b128 %0, %1, %2"
                     :
                     : "v"(lds_addr), "v"(goff), "s"(rowbase)
                     : "memory");
    }
    asm volatile("s_wait_asynccnt 0x0" ::: "memory");
    __syncthreads();

    float s1 = 0.0f, s2 = 0.0f;
#pragma unroll
    for (int k = 0; k < 4; ++k) {
        vfloat4 v = tile[tid + k * 256];
        s1 += v.x + v.y + v.z + v.w;
        s2 = fmaf(v.x, v.x, fmaf(v.y, v.y, fmaf(v.z, v.z, fmaf(v.w, v.w, s2))));
    }
    // wave32 butterfly reduction
#pragma unroll
    for (int m = 16; m > 0; m >>= 1) {
        s1 += __shfl_xor(s1, m, 32);
        s2 += __shfl_xor(s2, m, 32);
    }
    if ((tid & 31) == 0) { ps1[tid >> 5] = s1; ps2[tid >> 5] = s2; }
    __syncthreads();
    if (tid == 0) {
        float a = 0.0f, b = 0.0f;
#pragma unroll
        for (int w = 0; w < 8; ++w) { a += ps1[w]; b += ps2[w]; }
        stats[bt] = make_float2(a, b);
    }
}

// ---------------------------------------------------------------------------
// Kernel 2: per-batch double prefix scan over T (tiny: 2048 steps).
// In-place over stats[]: reads (s1, ssq), writes (mean, inv_std).
//   mean[t]  = cumsum(s1)[t] / ((t+1)*N)
//   q[t]     = ssq[t] - 2*mean[t]*s1[t] + N*mean[t]^2      (reference expands
//              (x - mean[t])^2 summed over this t's N elements)
//   var[t]   = cumsum(q)[t] / ((t+1)*N)
// One block of 256 threads per batch; each thread owns 8 consecutive t.
// ---------------------------------------------------------------------------
__global__ __launch_bounds__(256) void k_scan(float2* __restrict__ stats) {
    __shared__ float sb[256];
    const int b   = blockIdx.x;
    const int tid = threadIdx.x;
    float2* st = stats + (size_t)b * T_LEN;
    const int t0 = tid * 8;

    float raw1[8], rawq[8], p[8];
#pragma unroll
    for (int j = 0; j < 8; ++j) {
        float2 v = st[t0 + j];
        raw1[j] = v.x;
        rawq[j] = v.y;          // ssq for now
    }

    // ---- scan 1: cumulative s1 ----
    p[0] = raw1[0];
#pragma unroll
    for (int j = 1; j < 8; ++j) p[j] = p[j - 1] + raw1[j];
    float tot = p[7];
    sb[tid] = tot; __syncthreads();
    for (int off = 1; off < 256; off <<= 1) {
        float v   = sb[tid];
        float add = (tid >= off) ? sb[tid - off] : 0.0f;
        __syncthreads();
        sb[tid] = v + add;
        __syncthreads();
    }
    float excl1 = sb[tid] - tot;

    float mean[8];
#pragma unroll
    for (int j = 0; j < 8; ++j) {
        const int   t   = t0 + j;
        const float cnt = (float)(t + 1) * (float)N_RED;
        mean[j] = (excl1 + p[j]) / cnt;
        // q[t] = ssq - 2*m*s1 + N*m^2
        rawq[j] = fmaf((float)N_RED * mean[j], mean[j],
                       fmaf(-2.0f * mean[j], raw1[j], rawq[j]));
    }

    // ---- scan 2: cumulative q ----
    __syncthreads();            // protect sb reuse
    p[0] = rawq[0];
#pragma unroll
    for (int j = 1; j < 8; ++j) p[j] = p[j - 1] + rawq[j];
    tot = p[7];
    sb[tid] = tot; __syncthreads();
    for (int off = 1; off < 256; off <<= 1) {
        float v   = sb[tid];
        float add = (tid >= off) ? sb[tid - off] : 0.0f;
        __syncthreads();
        sb[tid] = v + add;
        __syncthreads();
    }
    float excl2 = sb[tid] - tot;

#pragma unroll
    for (int j = 0; j < 8; ++j) {
        const int   t   = t0 + j;
        const float cnt = (float)(t + 1) * (float)N_RED;
        const float var = (excl2 + p[j]) / cnt;
        st[t0 + j] = make_float2(mean[j], rsqrtf(var + EPS_GN));
    }
}

// ---------------------------------------------------------------------------
// Kernel 3: elementwise normalize.  Processed in REVERSE block order so the
// tail of x (most recently streamed through the 192 MB L2 by k_stats) is
// consumed before eviction.  Output is write-once -> non-temporal stores so
// the 256 MiB stream doesn't evict L2-resident x.
// ---------------------------------------------------------------------------
__global__ __launch_bounds__(256) void k_norm(const float* __restrict__ x,
                                              const float2* __restrict__ mi,
                                              const float* __restrict__ scale,
                                              float* __restrict__ out,
                                              int nblocks) {
    const int    blk = nblocks - 1 - (int)blockIdx.x;   // reversed order
    const size_t f4  = (size_t)blk * 256 + threadIdx.x; // float4 index
    const size_t e0  = f4 * 4;                          // element index
    const int    bt  = (int)(e0 >> 12);                 // / 4096

    const float2  ms = mi[bt];                          // (mean, inv_std)
    const vfloat4 v  = ((const vfloat4*)x)[f4];
    const vfloat4 sc = ((const vfloat4*)scale)[(e0 & 31u) >> 2]; // C=32, 4-aligned

    vfloat4 o;
    o.x = (v.x - ms.x) * ms.y * sc.x;
    o.y = (v.y - ms.x) * ms.y * sc.y;
    o.z = (v.z - ms.x) * ms.y * sc.z;
    o.w = (v.w - ms.x) * ms.y * sc.w;

    __builtin_nontemporal_store(o, (vfloat4*)out + f4);
}

// ---------------------------------------------------------------------------
extern "C" void kernel_launch(void* const* d_in, const int* in_sizes, int n_in,
                              void* d_out, int out_size, void* d_ws, size_t ws_size,
                              hipStream_t stream) {
    const float* x     = (const float*)d_in[0];
    const float* scale = (const float*)d_in[1];
    float*       out   = (float*)d_out;
    float2*      stats = (float2*)d_ws;   // B*T float2 = 128 KB

    const int nbt = B_LEN * T_LEN;                         // 16384 blocks
    k_stats<<<nbt, 256, 0, stream>>>(x, stats);
    k_scan <<<B_LEN, 256, 0, stream>>>(stats);

    const int total_f4   = (B_LEN * T_LEN * N_RED) / 4;    // 16,777,216
    const int norm_blocks = total_f4 / 256;                // 65,536
    k_norm <<<norm_blocks, 256, 0, stream>>>(x, stats, scale, out, norm_blocks);
}